// Sampler_34419867910302
// MI455X (gfx1250) — compile-verified
//
#include <hip/hip_runtime.h>
#include <math.h>

#define THREADS 256
#define VOCAB   128000
#define VITERS  (VOCAB / (THREADS * 4))   // 125 float4 chunks per lane, exact tile

typedef float vf4 __attribute__((ext_vector_type(4)));

// CDNA5 split dependency counter for async LDS loads (SOPP op 74)
#define WAIT_ASYNC(n) asm volatile("s_wait_asynccnt " #n ::: "memory")

// order-preserving float <-> uint transform (ascending)
__device__ __forceinline__ unsigned f2ord(float f) {
  unsigned u = __float_as_uint(f);
  return (u & 0x80000000u) ? ~u : (u | 0x80000000u);
}
__device__ __forceinline__ float ord2f(unsigned o) {
  return __uint_as_float((o & 0x80000000u) ? (o & 0x7fffffffu) : ~o);
}

// ---- wave32 shuffle primitives (no barriers) -------------------------------
__device__ __forceinline__ unsigned wave_sufsum_u(unsigned v) {
  const int lane = threadIdx.x & 31;
#pragma unroll
  for (int off = 1; off < 32; off <<= 1) {
    unsigned t = (unsigned)__shfl_down((int)v, off, 32);
    if (lane + off < 32) v += t;
  }
  return v;   // inclusive suffix sum over the wave's 32 values
}
__device__ __forceinline__ float wave_sufsum_f(float v) {
  const int lane = threadIdx.x & 31;
#pragma unroll
  for (int off = 1; off < 32; off <<= 1) {
    float t = __shfl_down(v, off, 32);
    if (lane + off < 32) v += t;
  }
  return v;
}

// ---- 2-barrier block reductions (8 waves) ----------------------------------
__device__ __forceinline__ float block_sum(float v, float* w8) {
  const int lane = threadIdx.x & 31, w = threadIdx.x >> 5;
#pragma unroll
  for (int off = 1; off < 32; off <<= 1) {
    float t = __shfl_down(v, off, 32);
    if (lane + off < 32) v += t;
  }
  if (lane == 0) w8[w] = v;
  __syncthreads();
  float tot = 0.f;
#pragma unroll
  for (int i = 0; i < 8; ++i) tot += w8[i];
  __syncthreads();
  return tot;
}
__device__ __forceinline__ float block_max(float v, float* w8) {
  const int lane = threadIdx.x & 31, w = threadIdx.x >> 5;
#pragma unroll
  for (int off = 1; off < 32; off <<= 1) {
    float t = __shfl_down(v, off, 32);
    if (lane + off < 32) v = fmaxf(v, t);
  }
  if (lane == 0) w8[w] = v;
  __syncthreads();
  float tot = w8[0];
#pragma unroll
  for (int i = 1; i < 8; ++i) tot = fmaxf(tot, w8[i]);
  __syncthreads();
  return tot;
}
__device__ __forceinline__ void block_argmax(float v, unsigned idx,
                                             float* w8f, unsigned* w8u,
                                             float& outv, unsigned& outi) {
  const int lane = threadIdx.x & 31, w = threadIdx.x >> 5;
#pragma unroll
  for (int off = 1; off < 32; off <<= 1) {
    float    tv = __shfl_down(v, off, 32);
    unsigned ti = (unsigned)__shfl_down((int)idx, off, 32);
    if (lane + off < 32) {
      if (tv > v || (tv == v && ti < idx)) { v = tv; idx = ti; }
    }
  }
  if (lane == 0) { w8f[w] = v; w8u[w] = idx; }
  __syncthreads();
  float bv = w8f[0]; unsigned bi = w8u[0];
#pragma unroll
  for (int i = 1; i < 8; ++i) {
    float tv = w8f[i]; unsigned ti = w8u[i];
    if (tv > bv || (tv == bv && ti < bi)) { bv = tv; bi = ti; }
  }
  __syncthreads();
  outv = bv; outi = bi;
}

__global__ void __launch_bounds__(THREADS)
sampler_fused(const float* __restrict__ logits,
              const float* __restrict__ topp,
              const int*   __restrict__ topk,
              const float* __restrict__ minp,
              const float* __restrict__ expq,
              float* __restrict__ out_logits,
              float* __restrict__ out_probs,
              float* __restrict__ out_sample)
{
  const int row = blockIdx.x;
  const int tid = threadIdx.x;
  const vf4* __restrict__ xr = (const vf4*)(logits + (size_t)row * VOCAB);
  const vf4* __restrict__ qr = (const vf4*)(expq   + (size_t)row * VOCAB);

  __shared__ vf4      stage[4][THREADS];   // 4-deep async-to-LDS pipeline (16 KB)
  __shared__ unsigned hist[THREADS];       // 256-bin count histogram
  __shared__ float    whist[THREADS];      // 256-bin exp-weight histogram
  __shared__ float    redf[THREADS];       // suffix sums (float rounds)
  __shared__ unsigned redu[THREADS];       // suffix sums (count rounds)
  __shared__ float    w8f[8];
  __shared__ unsigned w8u[8];
  __shared__ int      s_bin;

  //======== Pass 1 (the only HBM-bound logits pass): row max + radix round 0,
  //======== streamed via 4-deep global_load_async_to_lds_b128 / ASYNCcnt.
  hist[tid] = 0u;
  __syncthreads();

  const unsigned ldsbase = (unsigned)(unsigned long long)&stage[0][tid];
#pragma unroll
  for (int pq = 0; pq < 3; ++pq) {   // prologue: 3 chunks in flight
    unsigned long long ga = (unsigned long long)(xr + pq * THREADS + tid);
    unsigned dst = ldsbase + (unsigned)pq * (THREADS * 16u);
    asm volatile("global_load_async_to_lds_b128 %0, %1, off"
                 :: "v"(dst), "v"(ga) : "memory");
  }
  float lmax = -INFINITY;
  for (int it = 0; it < VITERS; ++it) {
    if (it + 3 < VITERS) {
      unsigned long long ga = (unsigned long long)(xr + (it + 3) * THREADS + tid);
      unsigned dst = ldsbase + (unsigned)((it + 3) & 3) * (THREADS * 16u);
      asm volatile("global_load_async_to_lds_b128 %0, %1, off"
                   :: "v"(dst), "v"(ga) : "memory");
      WAIT_ASYNC(3);            // oldest (chunk it) has landed in LDS
    } else {
      WAIT_ASYNC(0);            // drain tail
    }
    vf4 v = stage[it & 3][tid]; // each lane reads only its own 16B slot
#pragma unroll
    for (int j = 0; j < 4; ++j) {
      float e = v[j];
      lmax = fmaxf(lmax, e);
      atomicAdd(&hist[f2ord(e) >> 24], 1u);
    }
  }
  const float m = block_max(lmax, w8f);

  //======== Exact k-th largest logit via MSD radix select (L2-resident passes)
  int kk = topk[row];
  kk = kk < 1 ? 1 : (kk > VOCAB ? VOCAB : kk);
  unsigned pref = 0u;
  int rank = kk;
  for (int shift = 24; shift >= 0; shift -= 8) {
    if (shift != 24) {
      hist[tid] = 0u;
      __syncthreads();
      const unsigned pmask = 0xffffffffu << (shift + 8);
      for (int it = 0; it < VITERS; ++it) {
        vf4 v = xr[it * THREADS + tid];
#pragma unroll
        for (int j = 0; j < 4; ++j) {
          unsigned key = f2ord(v[j]);
          if ((key & pmask) == pref) atomicAdd(&hist[(key >> shift) & 0xffu], 1u);
        }
      }
      __syncthreads();
    }
    // barrier-light inclusive suffix sums of the 256 bins
    unsigned sufw = wave_sufsum_u(hist[tid]);
    if ((tid & 31) == 0) w8u[tid >> 5] = sufw;
    __syncthreads();
    unsigned addu = 0;
#pragma unroll
    for (int i = 0; i < 8; ++i) if (i > (tid >> 5)) addu += w8u[i];
    redu[tid] = sufw + addu;
    if (tid == 0) s_bin = 0;
    __syncthreads();
    const unsigned ur = (unsigned)rank;
    const unsigned sufb  = redu[tid];
    const unsigned sufb1 = (tid + 1 < THREADS) ? redu[tid + 1] : 0u;
    if (sufb >= ur && sufb1 < ur) s_bin = tid;   // unique winner (monotone)
    __syncthreads();
    const int b = s_bin;
    const unsigned above = (b + 1 < THREADS) ? redu[b + 1] : 0u;
    pref |= ((unsigned)b) << shift;
    rank -= (int)above;
    __syncthreads();
  }
  const float t_k = ord2f(pref);   // reference keeps x >= kth largest value

  //======== Pass: Z over top-k survivors + weighted radix round 0
  whist[tid] = 0.f;
  __syncthreads();
  float lz = 0.f;
  for (int it = 0; it < VITERS; ++it) {
    vf4 v = xr[it * THREADS + tid];
#pragma unroll
    for (int j = 0; j < 4; ++j) {
      float e = v[j];
      if (e >= t_k) {
        float w = __expf(e - m);
        lz += w;
        atomicAdd(&whist[f2ord(e) >> 24], w);   // ds_add_f32 LDS float atomics
      }
    }
  }
  const float Z = block_sum(lz, w8f);

  //======== top-p value threshold v*: token where descending cumulative
  //======== exp-weight first reaches p*Z (weighted radix select)
  float rem = topp[row] * Z;
  unsigned wpref = 0u;
  for (int shift = 24; shift >= 0; shift -= 8) {
    if (shift != 24) {
      whist[tid] = 0.f;
      __syncthreads();
      const unsigned pmask = 0xffffffffu << (shift + 8);
      for (int it = 0; it < VITERS; ++it) {
        vf4 v = xr[it * THREADS + tid];
#pragma unroll
        for (int j = 0; j < 4; ++j) {
          float e = v[j];
          if (e >= t_k) {
            unsigned key = f2ord(e);
            if ((key & pmask) == wpref)
              atomicAdd(&whist[(key >> shift) & 0xffu], __expf(e - m));
          }
        }
      }
      __syncthreads();
    }
    float sufw = wave_sufsum_f(whist[tid]);
    if ((tid & 31) == 0) w8f[tid >> 5] = sufw;
    __syncthreads();
    float addf = 0.f;
#pragma unroll
    for (int i = 0; i < 8; ++i) if (i > (tid >> 5)) addf += w8f[i];
    redf[tid] = sufw + addf;
    if (tid == 0) s_bin = 0;   // degenerate (p~1) -> falls through to keep-all
    __syncthreads();
    const float sufb  = redf[tid];
    const float sufb1 = (tid + 1 < THREADS) ? redf[tid + 1] : 0.f;
    if (sufb >= rem && sufb1 < rem) s_bin = tid;
    __syncthreads();
    const int b = s_bin;
    const float above = (b + 1 < THREADS) ? redf[b + 1] : 0.f;
    wpref |= ((unsigned)b) << shift;
    rem -= above;
    __syncthreads();
  }
  const float vstar = ord2f(wpref);   // NaN in keep-all case; fmaxf drops it

  //======== min-p threshold (closed form) + combined final threshold
  const float t_mp = m + __logf(minp[row]);       // min_p==0 -> -inf -> no-op
  float T = fmaxf(t_k, t_mp);
  T = fmaxf(T, vstar);

  //======== Pass: Z3 over final survivors + exponential-trick argmax
  float lz3 = 0.f;
  float lbest = -INFINITY;
  unsigned lbi = 0xffffffffu;
  for (int it = 0; it < VITERS; ++it) {
    const int base = it * THREADS + tid;
    vf4 v  = xr[base];
    vf4 qv = __builtin_nontemporal_load(&qr[base]);   // single-use: NT, spare L2
#pragma unroll
    for (int j = 0; j < 4; ++j) {
      float e = v[j];
      if (e >= T) {
        lz3 += __expf(e - m);
        float sc = e - __logf(qv[j]);   // argmax probs/q == argmax (x - log q)
        unsigned gi = (unsigned)(base * 4 + j);
        if (sc > lbest || (sc == lbest && gi < lbi)) { lbest = sc; lbi = gi; }
      }
    }
  }
  const float Z3 = block_sum(lz3, w8f);
  float bestv; unsigned besti;
  block_argmax(lbest, lbi, w8f, w8u, bestv, besti);
  if (tid == 0) out_sample[row] = (float)besti;

  //======== Final pass: write logits_f and probs with streaming NT stores
  const float invZ3 = 1.0f / Z3;
  vf4* __restrict__ ol = (vf4*)(out_logits + (size_t)row * VOCAB);
  vf4* __restrict__ op = (vf4*)(out_probs  + (size_t)row * VOCAB);
  for (int it = 0; it < VITERS; ++it) {
    const int base = it * THREADS + tid;
    vf4 v = xr[base];
    vf4 lo = v, pr = v;
#pragma unroll
    for (int j = 0; j < 4; ++j) {
      float e = v[j];
      bool kept = (e >= T);
      lo[j] = kept ? e : -INFINITY;
      pr[j] = kept ? __expf(e - m) * invZ3 : 0.f;
    }
    __builtin_nontemporal_store(lo, &ol[base]);
    __builtin_nontemporal_store(pr, &op[base]);
  }
}

extern "C" void kernel_launch(void* const* d_in, const int* in_sizes, int n_in,
                              void* d_out, int out_size, void* d_ws, size_t ws_size,
                              hipStream_t stream) {
  (void)n_in; (void)out_size; (void)d_ws; (void)ws_size;
  const float* logits = (const float*)d_in[0];
  const float* p      = (const float*)d_in[1];
  const int*   k      = (const int*)d_in[2];
  const float* minp   = (const float*)d_in[3];
  const float* q      = (const float*)d_in[4];
  const int B = in_sizes[1];                 // 256 rows
  float* out_logits = (float*)d_out;
  float* out_probs  = out_logits + (size_t)B * VOCAB;
  float* out_sample = out_probs  + (size_t)B * VOCAB;
  sampler_fused<<<B, THREADS, 0, stream>>>(logits, p, k, minp, q,
                                           out_logits, out_probs, out_sample);
}